// CRFEntityLayer_13331578487515
// MI455X (gfx1250) — compile-verified
//
#include <hip/hip_runtime.h>
#include <hip/hip_bf16.h>

typedef __attribute__((ext_vector_type(2))) float v2f;
typedef __attribute__((ext_vector_type(8))) float v8f;

#define B_ 64
#define T_ 512
#define H_ 768
#define K_ 32

// ---------------------------------------------------------------------------
// Kernel 1: pot[b,t,k] = x[b,t,:] @ W[:,k] + b[k]   via V_WMMA_F32_16X16X4_F32
// One wave = one 16-row x 32-col tile (two 16x16 accumulators), 192 k-steps.
// A layout (ISA 7.12.2, 32b A 16x4): lanes 0-15 -> k={0,1}, lanes 16-31 -> k={2,3}
// B/C/D: row striped across lanes; lane col = lane&15, half select = lane>>4.
// ---------------------------------------------------------------------------
__global__ __launch_bounds__(256)
void gemm_pot_wmma(const float* __restrict__ x,
                   const float* __restrict__ W,
                   const float* __restrict__ bias,
                   float* __restrict__ pot) {
  const int lane    = threadIdx.x & 31;
  const int wave    = blockIdx.x * (blockDim.x >> 5) + (threadIdx.x >> 5);
  const int m0      = wave * 16;            // 2048 waves * 16 rows = B_*T_
  const int halfsel = lane >> 4;
  const int l15     = lane & 15;
  const int kb      = 2 * halfsel;

  const float* arow = x + (size_t)(m0 + l15) * H_ + kb;

  v8f c0, c1;
  const float b0v = bias[l15];
  const float b1v = bias[16 + l15];
#pragma unroll
  for (int r = 0; r < 8; ++r) { c0[r] = b0v; c1[r] = b1v; }

#pragma unroll 4
  for (int h = 0; h < H_; h += 4) {
    v2f a = *(const v2f*)(arow + h);        // contiguous k-pair, 8B aligned
    const float* wrow = W + (size_t)(h + kb) * K_;
    v2f b0, b1;
    b0.x = wrow[l15];        b0.y = wrow[K_ + l15];
    b1.x = wrow[16 + l15];   b1.y = wrow[K_ + 16 + l15];
    c0 = __builtin_amdgcn_wmma_f32_16x16x4_f32(false, a, false, b0, (short)0, c0, false, false);
    c1 = __builtin_amdgcn_wmma_f32_16x16x4_f32(false, a, false, b1, (short)0, c1, false, false);
  }

#pragma unroll
  for (int r = 0; r < 8; ++r) {
    const size_t row = (size_t)(m0 + 8 * halfsel + r);
    pot[row * K_ + l15]      = c0[r];
    pot[row * K_ + 16 + l15] = c1[r];
  }
}

// ---------------------------------------------------------------------------
// Kernel 2: CRF forward logsumexp + Viterbi + sequence score + backtrack.
// One wave per batch element; lane j owns tag j; trans column in registers.
// ---------------------------------------------------------------------------
__global__ __launch_bounds__(32)
void crf_scan(const float* __restrict__ pot,
              const int*   __restrict__ y,
              const int*   __restrict__ lens,
              const float* __restrict__ trans,
              float*       __restrict__ out) {
  __shared__ unsigned char bp[T_ * K_];     // 16 KB backpointers

  const int b    = blockIdx.x;
  const int lane = threadIdx.x;
  const int len  = lens[b];
  const float* pb = pot + (size_t)b * T_ * K_;
  const int*   yb = y   + (size_t)b * T_;

  float tc[K_];                              // trans[i][lane], i = 0..31
#pragma unroll
  for (int i = 0; i < K_; ++i) tc[i] = trans[i * K_ + lane];

  float alpha_l = pb[lane];                  // logsumexp alpha
  float alpha_v = alpha_l;                   // viterbi alpha

  int   y_prev = yb[0];
  float score  = __shfl(alpha_l, y_prev);    // unary at t=0 (always < len)

  for (int t = 1; t < T_; ++t) {
    const float p    = pb[t * K_ + lane];
    const bool  keep = (t < len);

    // Pass 1: broadcasts + max / argmax (short dep chain, readlane-friendly)
    float vl[K_];
    float m_l = -__builtin_inff();
    float m_v = -__builtin_inff();
    int   arg = 0;
#pragma unroll
    for (int i = 0; i < K_; ++i) {
      const float al = __shfl(alpha_l, i);
      const float av = __shfl(alpha_v, i);
      vl[i] = al + tc[i];
      m_l   = fmaxf(m_l, vl[i]);
      const float vv = av + tc[i];
      if (vv > m_v) { m_v = vv; arg = i; }   // strict > keeps first max (argmax)
    }

    // Pass 2: 32 independent v_exp_f32 into 4 partial sums (TRANS co-exec)
    float s0 = 0.f, s1 = 0.f, s2 = 0.f, s3 = 0.f;
#pragma unroll
    for (int i = 0; i < K_; i += 4) {
      s0 += __expf(vl[i]     - m_l);
      s1 += __expf(vl[i + 1] - m_l);
      s2 += __expf(vl[i + 2] - m_l);
      s3 += __expf(vl[i + 3] - m_l);
    }
    const float new_l = m_l + __logf((s0 + s1) + (s2 + s3)) + p;
    const float new_v = m_v + p;

    const int y_t = yb[t];
    if (keep) {                              // uniform branch across the wave
      alpha_l = new_l;
      alpha_v = new_v;
      // unary mask (t<len) and binary mask (t-1 < len-1) coincide with keep
      score += __shfl(p, y_t) + trans[y_prev * K_ + y_t];
    }
    bp[t * K_ + lane] = (unsigned char)(keep ? arg : lane);  // identity if frozen
    y_prev = y_t;
  }

  // log_norm = logsumexp over lanes of alpha_l
  float m = alpha_l;
#pragma unroll
  for (int off = 16; off; off >>= 1) m = fmaxf(m, __shfl_xor(m, off));
  float e = __expf(alpha_l - m);
#pragma unroll
  for (int off = 16; off; off >>= 1) e += __shfl_xor(e, off);
  const float log_norm = m + __logf(e);

  // last = argmax over lanes of alpha_v (ties -> lowest index, like jnp.argmax)
  float v   = alpha_v;
  int   idx = lane;
#pragma unroll
  for (int off = 16; off; off >>= 1) {
    const float ov = __shfl_xor(v, off);
    const int   oi = __shfl_xor(idx, off);
    if (ov > v || (ov == v && oi < idx)) { v = ov; idx = oi; }
  }

  __syncthreads();                           // make bp[] visible / drain DScnt

  if (lane == 0) {
    out[b] = score - log_norm;               // log-likelihood
    float* preds = out + B_ + (size_t)b * T_;
    int tag = idx;
    for (int t = T_ - 1; t >= 1; --t) {
      preds[t] = (t < len) ? (float)tag : 0.f;   // preds * mask
      tag = bp[t * K_ + tag];
    }
    preds[0] = (float)tag;                   // t=0 always inside sequence
  }
}

// ---------------------------------------------------------------------------
// d_in: x[64*512*768] f32, y[64*512] i32, lens[64] i32, W[768*32] f32,
//       b[32] f32, trans[32*32] f32
// d_out: 64 f32 log-likelihood  ++  64*512 predictions (written as f32)
// d_ws : pot scratch, 64*512*32 floats = 4 MB
// ---------------------------------------------------------------------------
extern "C" void kernel_launch(void* const* d_in, const int* in_sizes, int n_in,
                              void* d_out, int out_size, void* d_ws, size_t ws_size,
                              hipStream_t stream) {
  (void)in_sizes; (void)n_in; (void)out_size; (void)ws_size;
  const float* x     = (const float*)d_in[0];
  const int*   y     = (const int*)d_in[1];
  const int*   lens  = (const int*)d_in[2];
  const float* W     = (const float*)d_in[3];
  const float* bias  = (const float*)d_in[4];
  const float* trans = (const float*)d_in[5];
  float* out = (float*)d_out;
  float* pot = (float*)d_ws;

  // (B_*T_)/16 = 2048 tiles, 8 waves/block -> 256 blocks
  gemm_pot_wmma<<<256, 256, 0, stream>>>(x, W, bias, pot);
  crf_scan<<<B_, 32, 0, stream>>>(pot, y, lens, trans, out);
}